// HGTEncoder_36352603193978
// MI455X (gfx1250) — compile-verified
//
#include <hip/hip_runtime.h>
#include <hip/hip_bf16.h>
#include <math.h>

// ---------------- problem constants (match reference) ----------------
#define NN 30000     // nodes per type
#define NPAD 30016   // 32 * 938 : M padded to the 32-row block tile
#define EE 200000    // edges per edge type
#define FF 256       // hidden dim (= K of every GEMM)
#define HH 8         // heads
#define DD 32        // per-head dim
#define LL 3         // layers
#define F3 (3 * FF)  // 768

#define LDS_STRIDE 260  // 256 + 4 pad -> conflict-free strided reads

typedef __attribute__((ext_vector_type(2))) float v2f;
typedef __attribute__((ext_vector_type(8))) float v8f;

// ---------------- small helpers ----------------
__device__ inline void atomicMaxF(float* addr, float val) {
    unsigned int* ua = (unsigned int*)addr;
    unsigned int old = *ua;
    while (__uint_as_float(old) < val) {
        unsigned int assumed = old;
        old = atomicCAS(ua, assumed, __float_as_uint(val));
        if (old == assumed) break;
    }
}

__global__ void fill_kernel(float* __restrict__ p, float v, int n) {
    int i = blockIdx.x * blockDim.x + threadIdx.x;
    if (i < n) p[i] = v;
}

// ---------------- fused-weight prep ----------------
// Wf[l][t] : [FF][3F] row-major.  cols [0,F): Wk-block @ A_bd, scaled by p/sqrt(D)
//                                 cols [F,2F): Wq copied
//                                 cols [2F,3F): Wv-block @ M_bd
// row == FF encodes the fused bias row.
__global__ void prep_fused_kernel(const float* __restrict__ W_kqv,
                                  const float* __restrict__ b_kqv,
                                  const float* __restrict__ A_rel,
                                  const float* __restrict__ M_rel,
                                  const float* __restrict__ p_rel,
                                  float* __restrict__ Wf,
                                  float* __restrict__ bf) {
    const int rows = FF + 1;
    long idx = (long)blockIdx.x * blockDim.x + threadIdx.x;
    long total = (long)LL * 2 * rows * F3;
    if (idx >= total) return;
    int col = (int)(idx % F3);
    long r1 = idx / F3;
    int row = (int)(r1 % rows);
    int lt = (int)(r1 / rows);           // l*2 + t

    const float* Wsrc = W_kqv + (long)lt * FF * F3;   // [FF][3F]
    const float* bsrc = b_kqv + (long)lt * F3;

    float out;
    if (col >= FF && col < 2 * FF) {                  // q passthrough
        out = (row == FF) ? bsrc[col] : Wsrc[(long)row * F3 + col];
    } else {
        const bool isK = (col < FF);
        const int j = isK ? col : (col - 2 * FF);
        const int h = j >> 5;                         // /DD
        const int eo = j & 31;                        // %DD
        const float* T = (isK ? A_rel : M_rel) + ((long)(lt * HH + h)) * DD * DD;
        const int base = isK ? 0 : 2 * FF;
        float s = 0.f;
#pragma unroll
        for (int d = 0; d < DD; ++d) {
            float w = (row == FF) ? bsrc[base + h * DD + d]
                                  : Wsrc[(long)row * F3 + base + h * DD + d];
            s += w * T[d * DD + eo];
        }
        if (isK) s *= p_rel[lt * HH + h] * 0.17677669529663687f; // 1/sqrt(32)
        out = s;
    }
    if (row == FF) bf[(long)lt * F3 + col] = out;
    else           Wf[((long)lt * FF + row) * F3 + col] = out;
}

// ---------------- B packing for WMMA lane layout ----------------
// src: `cnt` matrices of [FF rows (K) x Nout] row-major, back to back.
// dst float2 layout per matrix: pair[(kg*2 + lh)*Nout + c] = {B[4kg+2lh][c], B[4kg+2lh+1][c]}
// -> in the GEMM each lane fetches its (b0,b1) operand pair with ONE global_load_b64.
__global__ void pack_b_kernel(const float* __restrict__ src, float* __restrict__ dst,
                              int Nout, int cnt) {
    long idx = (long)blockIdx.x * blockDim.x + threadIdx.x;
    long per = (long)FF * Nout;
    long total = per * cnt;
    if (idx >= total) return;
    int m = (int)(idx / per);
    long rc = idx % per;
    int r = (int)(rc / Nout);
    int c = (int)(rc % Nout);
    long o2 = (((long)(r >> 2) * 2 + ((r >> 1) & 1)) * Nout + c);  // float2 index
    dst[m * per + o2 * 2 + (r & 1)] = src[idx];
}

// ---------------- WMMA fp32 GEMM: C = A[Mx256] * B[256xN] + bias ----------------
// block = 128 threads = 4 waves; block tile 32 rows x 256 cols (wave: 32x64, 8x v8f acc).
// A tile (32x256 = one contiguous 32KB span, M padded to NPAD) staged via LDS.
// B pre-packed (see pack_b_kernel): one b64 per lane feeds TWO WMMAs (row sub-tiles).
// Stores guarded by row < NN (pad rows never escape).
// mode 0: C = acc + bias
// mode 1: C = relu( g*(acc+bias) + (1-g)*xprev ),  g = sigmoid(*skipp)
__global__ __launch_bounds__(128) void wmma_gemm_kernel(
        const float* __restrict__ A, const float2* __restrict__ Bp,
        const float* __restrict__ bias, float* __restrict__ C,
        int Nout, int mode,
        const float* __restrict__ xprev, const float* __restrict__ skipp) {
    __shared__ float As[32 * LDS_STRIDE];

    const int row0 = blockIdx.x << 5;

    // cooperative A stage: source tile is contiguous (lda == 256, rows < NPAD)
    {
        const float4* src = (const float4*)(A + (size_t)row0 * FF);
#pragma unroll
        for (int i = threadIdx.x; i < 32 * 64; i += 128) {
            float4 v = src[i];
            *(float4*)(As + (i >> 6) * LDS_STRIDE + ((i & 63) << 2)) = v;
        }
    }
    __syncthreads();

    const int wave = threadIdx.x >> 5;
    const int lane = threadIdx.x & 31;
    const int lm = lane & 15;
    const int lh = lane >> 4;
    const int col0 = (blockIdx.y * 4 + wave) * 64;   // grid.y = Nout/256 exactly

    v8f acc[2][4] = {};
    const float* arow0 = As + lm * LDS_STRIDE + 2 * lh;         // rows 0..15 of tile
    const float* arow1 = As + (lm + 16) * LDS_STRIDE + 2 * lh;  // rows 16..31 of tile

#pragma unroll 2
    for (int kg = 0; kg < FF / 4; ++kg) {
        const float2 a0 = *(const float2*)(arow0 + kg * 4);     // ds_load_b64
        const float2 a1 = *(const float2*)(arow1 + kg * 4);     // ds_load_b64
        const float2* brow = Bp + (size_t)(kg * 2 + lh) * Nout + col0 + lm;
        v2f va0; va0[0] = a0.x; va0[1] = a0.y;
        v2f va1; va1[0] = a1.x; va1[1] = a1.y;
#pragma unroll
        for (int j = 0; j < 4; ++j) {
            const float2 b2 = brow[j * 16];                     // global_load_b64
            v2f b; b[0] = b2.x; b[1] = b2.y;
            acc[0][j] = __builtin_amdgcn_wmma_f32_16x16x4_f32(
                false, va0, false, b, (short)0, acc[0][j], false, false);
            acc[1][j] = __builtin_amdgcn_wmma_f32_16x16x4_f32(
                false, va1, false, b, (short)0, acc[1][j], false, false);
        }
    }

    float g = 0.f;
    if (mode == 1) {
        float sv = *skipp;
        g = 1.0f / (1.0f + __expf(-sv));
    }
#pragma unroll
    for (int j = 0; j < 4; ++j) {
        const int col = col0 + j * 16 + lm;
        const float bb = bias[col];
#pragma unroll
        for (int m = 0; m < 2; ++m) {
#pragma unroll
            for (int r = 0; r < 8; ++r) {
                const int row = row0 + m * 16 + r + 8 * lh;
                if (row < NN) {
                    float val = acc[m][j][r] + bb;
                    if (mode == 1) {
                        float xp = xprev[(size_t)row * Nout + col];
                        val = g * val + (1.0f - g) * xp;
                        val = val > 0.f ? val : 0.f;
                    }
                    C[(size_t)row * Nout + col] = val;
                }
            }
        }
    }
}

// ---------------- edge attention ----------------
// kqv layout per node: [ke'(F) | q(F) | ve'(F)] ; ke' already scaled by p/sqrt(D)
__global__ void edge_score_max_kernel(const float* __restrict__ kqv_src,
                                      const float* __restrict__ kqv_dst,
                                      const int* __restrict__ edges,
                                      float* __restrict__ score,
                                      float* __restrict__ mbuf, int nE) {
    int idx = blockIdx.x * blockDim.x + threadIdx.x;
    if (idx >= nE * HH) return;
    int e = idx >> 3, h = idx & 7;
    int src = edges[e];
    int dst = edges[nE + e];
    const float* qp = kqv_dst + (long)dst * F3 + FF + h * DD;
    const float* kp = kqv_src + (long)src * F3 + h * DD;
    float s = 0.f;
#pragma unroll
    for (int i = 0; i < 8; ++i) {
        float4 a = *(const float4*)(qp + 4 * i);
        float4 b = *(const float4*)(kp + 4 * i);
        s += a.x * b.x + a.y * b.y + a.z * b.z + a.w * b.w;
    }
    score[idx] = s;
    atomicMaxF(mbuf + (long)dst * HH + h, s);
}

__global__ void edge_exp_sum_kernel(const int* __restrict__ edges,
                                    float* __restrict__ score,
                                    const float* __restrict__ mbuf,
                                    float* __restrict__ zbuf, int nE) {
    int idx = blockIdx.x * blockDim.x + threadIdx.x;
    if (idx >= nE * HH) return;
    int e = idx >> 3, h = idx & 7;
    int dst = edges[nE + e];
    float w = __expf(score[idx] - mbuf[(long)dst * HH + h]);
    score[idx] = w;
    atomicAdd(zbuf + (long)dst * HH + h, w);
}

__global__ void edge_scatter_kernel(const float* __restrict__ kqv_src,
                                    const int* __restrict__ edges,
                                    const float* __restrict__ score,
                                    const float* __restrict__ zbuf,
                                    float* __restrict__ agg_dst, int nE) {
    int idx = blockIdx.x * blockDim.x + threadIdx.x;
    if (idx >= nE * HH) return;
    int e = idx >> 3, h = idx & 7;
    int src = edges[e];
    int dst = edges[nE + e];
    float alpha = score[idx] / zbuf[(long)dst * HH + h];
    const float* vp = kqv_src + (long)src * F3 + 2 * FF + h * DD;
    float* ap = agg_dst + (long)dst * FF + h * DD;
#pragma unroll
    for (int i = 0; i < 8; ++i) {
        float4 v4 = *(const float4*)(vp + 4 * i);
        atomicAdd(ap + 4 * i + 0, alpha * v4.x);
        atomicAdd(ap + 4 * i + 1, alpha * v4.y);
        atomicAdd(ap + 4 * i + 2, alpha * v4.z);
        atomicAdd(ap + 4 * i + 3, alpha * v4.w);
    }
}

__global__ void gelu_kernel(float* __restrict__ x, int n) {
    int i = blockIdx.x * blockDim.x + threadIdx.x;
    if (i >= n) return;
    float v = x[i];
    x[i] = 0.5f * v * (1.0f + erff(v * 0.70710678118654752f)); // exact gelu
}

// ---------------- host orchestration ----------------
extern "C" void kernel_launch(void* const* d_in, const int* in_sizes, int n_in,
                              void* d_out, int out_size, void* d_ws, size_t ws_size,
                              hipStream_t stream) {
    (void)in_sizes; (void)n_in; (void)out_size; (void)ws_size;
    const float* x_drug  = (const float*)d_in[0];
    const float* x_react = (const float*)d_in[1];
    const int*   edge_dr = (const int*)d_in[2];
    const int*   edge_rd = (const int*)d_in[3];
    const float* W_kqv   = (const float*)d_in[4];
    const float* b_kqv   = (const float*)d_in[5];
    const float* A_rel   = (const float*)d_in[6];
    const float* M_rel   = (const float*)d_in[7];
    const float* p_rel   = (const float*)d_in[8];
    const float* W_out   = (const float*)d_in[9];
    const float* b_out   = (const float*)d_in[10];
    const float* skip    = (const float*)d_in[11];
    float* out = (float*)d_out;

    // workspace carve-up (256B aligned)
    char* ws = (char*)d_ws;
    size_t off = 0;
    auto carve = [&](size_t bytes) -> char* {
        char* p = ws + off;
        off += (bytes + 255) & ~(size_t)255;
        return p;
    };
    float* Wf    = (float*)carve((size_t)LL * 2 * FF * F3 * 4);    // fused, unpacked
    float* Wfp   = (float*)carve((size_t)LL * 2 * FF * F3 * 4);    // fused, WMMA-packed
    float* Wop   = (float*)carve((size_t)LL * 2 * FF * FF * 4);    // W_out, WMMA-packed
    float* bf    = (float*)carve((size_t)LL * 2 * F3 * 4);
    float* kqv   = (float*)carve((size_t)2 * NN * F3 * 4);         // per type (GEMM out, guarded)
    float* xIn   = (float*)carve((size_t)2 * NPAD * FF * 4);       // padded copies of inputs
    float* xA    = (float*)carve((size_t)2 * NPAD * FF * 4);       // ping (padded)
    float* xB    = (float*)carve((size_t)2 * NPAD * FF * 4);       // pong (padded)
    float* agg   = (float*)carve((size_t)2 * NPAD * FF * 4);       // padded
    float* score = (float*)carve((size_t)EE * HH * 4);
    float* mb    = (float*)carve((size_t)2 * NN * HH * 4);
    float* zb    = (float*)carve((size_t)2 * NN * HH * 4);

    // 0) stage node features into padded A buffers (d2d async copies are capture-safe)
    hipMemcpyAsync(xIn,                       x_drug,  (size_t)NN * FF * 4,
                   hipMemcpyDeviceToDevice, stream);
    hipMemcpyAsync(xIn + (size_t)NPAD * FF,   x_react, (size_t)NN * FF * 4,
                   hipMemcpyDeviceToDevice, stream);

    // 1) fuse relation transforms + p/sqrt(D) into kqv weights (all layers), then
    //    pack both weight families into the WMMA b64 lane layout.
    {
        long total = (long)LL * 2 * (FF + 1) * F3;
        prep_fused_kernel<<<(int)((total + 255) / 256), 256, 0, stream>>>(
            W_kqv, b_kqv, A_rel, M_rel, p_rel, Wf, bf);
        long tp1 = (long)LL * 2 * FF * F3;
        pack_b_kernel<<<(int)((tp1 + 255) / 256), 256, 0, stream>>>(Wf, Wfp, F3, LL * 2);
        long tp2 = (long)LL * 2 * FF * FF;
        pack_b_kernel<<<(int)((tp2 + 255) / 256), 256, 0, stream>>>(W_out, Wop, FF, LL * 2);
    }

    const int mtiles = NPAD / 32;          // 938, exact on padded M
    const int eth = (EE * HH + 255) / 256; // edge-thread blocks

    const float* xcur[2] = {xIn, xIn + (size_t)NPAD * FF};
    float* pp[2] = {xA, xB};

    for (int l = 0; l < LL; ++l) {
        // 2) fused kqv GEMM per type -> [ke' | q | ve']
        for (int t = 0; t < 2; ++t) {
            int lt = l * 2 + t;
            dim3 grid(mtiles, F3 / 256);
            wmma_gemm_kernel<<<grid, 128, 0, stream>>>(
                xcur[t], (const float2*)(Wfp + (size_t)lt * FF * F3),
                bf + (size_t)lt * F3, kqv + (size_t)t * NN * F3,
                F3, 0, nullptr, nullptr);
        }

        // 3) init softmax state + aggregation (pads zeroed too)
        fill_kernel<<<(2 * NN * HH + 255) / 256, 256, 0, stream>>>(mb, -3.0e38f, 2 * NN * HH);
        fill_kernel<<<(2 * NN * HH + 255) / 256, 256, 0, stream>>>(zb, 0.0f, 2 * NN * HH);
        fill_kernel<<<(2 * NPAD * FF + 255) / 256, 256, 0, stream>>>(agg, 0.0f, 2 * NPAD * FF);

        // 4) edge attention per edge type (e=0: drug->react, e=1: react->drug)
        for (int e = 0; e < 2; ++e) {
            const int st = e, dt = 1 - e;
            const int* ei = (e == 0) ? edge_dr : edge_rd;
            const float* ks = kqv + (size_t)st * NN * F3;
            const float* kd = kqv + (size_t)dt * NN * F3;
            float* mdt = mb + (size_t)dt * NN * HH;
            float* zdt = zb + (size_t)dt * NN * HH;
            edge_score_max_kernel<<<eth, 256, 0, stream>>>(ks, kd, ei, score, mdt, EE);
            edge_exp_sum_kernel<<<eth, 256, 0, stream>>>(ei, score, mdt, zdt, EE);
            edge_scatter_kernel<<<eth, 256, 0, stream>>>(ks, ei, score, zdt,
                                                         agg + (size_t)dt * NPAD * FF, EE);
        }

        // 5) exact GELU in place on aggregated messages (both types; pads are zero)
        gelu_kernel<<<(2 * NPAD * FF + 255) / 256, 256, 0, stream>>>(agg, 2 * NPAD * FF);

        // 6) output GEMM with fused bias + sigmoid-skip + relu epilogue
        const bool last = (l == LL - 1);
        float* xnext_base = last ? out : pp[l & 1];
        const size_t tstride = last ? (size_t)NN * FF : (size_t)NPAD * FF;
        for (int t = 0; t < 2; ++t) {
            int lt = l * 2 + t;
            dim3 grid(mtiles, FF / 256);
            wmma_gemm_kernel<<<grid, 128, 0, stream>>>(
                agg + (size_t)t * NPAD * FF, (const float2*)(Wop + (size_t)lt * FF * FF),
                b_out + (size_t)lt * FF, xnext_base + t * tstride,
                FF, 1, xcur[t], skip + lt);
        }
        xcur[0] = xnext_base;
        xcur[1] = xnext_base + tstride;
    }
}